// HermiteEncoding_7670811591194
// MI455X (gfx1250) — compile-verified
//
#include <hip/hip_runtime.h>
#include <math.h>

// CDNA5 / gfx1250, wave32.
typedef __attribute__((ext_vector_type(2))) float v2f;
typedef __attribute__((ext_vector_type(8))) float v8f;

#define NPTS   65536
#define KOUT   8
#define NBODY  129      // HIDDEN + 1
#define HHALF  64
#define BLK    128      // points per block == threads per block (4 waves)

__global__ __launch_bounds__(BLK)
void hermite_fused(const float* __restrict__ e,
                   const float* __restrict__ W,
                   const float* __restrict__ b,
                   const float* __restrict__ scale_ptr,
                   float* __restrict__ out)
{
    __shared__ float F[BLK][25];   // 24 features per point, padded to 25 (bank-conflict free)
    __shared__ float Wc[24][16];   // collapsed weights: K=24, N padded 8->16 with zeros

    const int tid = threadIdx.x;
    const float scale = scale_ptr[0];

    // ---- zero the padded collapsed-weight tile ----
    for (int k = tid; k < 24 * 16; k += BLK) ((float*)Wc)[k] = 0.0f;
    __syncthreads();

    // ---- collapse each head's 129 weights to rank 3 (c0, S1, 2*S2) ----
    if (tid < KOUT) {
        const float* w = W + tid * NBODY;
        float s1 = 0.0f, s2 = 0.0f;
        for (int j = 1; j <= HHALF; ++j) {
            s1 += w[j];                      // weight of H0*gw columns
            s2 += (float)j * w[HHALF + j];   // weight of H1*gw columns: 2*t*j*gw
        }
        Wc[3 * tid + 0][tid] = w[0] + b[tid];
        Wc[3 * tid + 1][tid] = s1;
        Wc[3 * tid + 2][tid] = 2.0f * s2;
    }

    // ---- per-point feature phase: 8 heads x 64 exp terms ----
    const int ng = blockIdx.x * BLK + tid;
    const float ev = e[ng];
    float p = ev;                            // e^(i+1) running power
    for (int i = 0; i < KOUT; ++i) {
        const float t = scale * p;
        p *= ev;
        const float u = -0.5f * t * t;
        float s = 0.0f;
        #pragma unroll 8
        for (int j = 1; j <= HHALF; ++j) {
            const float jf = (float)j;
            s += expf(u * jf * jf);          // exp(-0.5 * (t*j)^2)
        }
        const float gw = s * (1.0f / 64.0f);
        F[tid][3 * i + 0] = 1.0f;
        F[tid][3 * i + 1] = gw;
        F[tid][3 * i + 2] = t * gw;
    }
    __syncthreads();

    // ---- projection: out(16x16) = F_tile(16x24) @ Wc(24x16) via f32 WMMA ----
    const int lane = tid & 31;
    const int wv   = tid >> 5;        // wave id: 16-point tile per wave
    const int m16  = lane & 15;
    const int half = lane >> 4;       // 0: K+{0,1}/rows 0-7, 1: K+{2,3}/rows 8-15

    v8f acc = {};
    #pragma unroll
    for (int kk = 0; kk < 24; kk += 4) {
        const int kA = kk + 2 * half;
        v2f a, bb;
        // A 16x4 f32: lanes0-15 -> K kk+0/kk+1 ; lanes16-31 -> K kk+2/kk+3 ; M = lane%16
        a.x = F[wv * 16 + m16][kA];
        a.y = F[wv * 16 + m16][kA + 1];
        // B 4x16 f32: VGPR0 -> K kk+2*half ; VGPR1 -> K kk+2*half+1 ; N = lane%16
        bb.x = Wc[kA][m16];
        bb.y = Wc[kA + 1][m16];
        // (neg_a, A, neg_b, B, c_mod, C, reuse_a, reuse_b)
        acc = __builtin_amdgcn_wmma_f32_16x16x4_f32(false, a, false, bb,
                                                    (short)0, acc, false, false);
    }

    // ---- D layout: VGPR r -> row (r + 8*half), col = lane%16. cols 8..15 are padding. ----
    if (m16 < KOUT) {
        float* op = out + m16 * NPTS + blockIdx.x * BLK + wv * 16 + 8 * half;
        float4 lo = make_float4(acc[0], acc[1], acc[2], acc[3]);
        float4 hi = make_float4(acc[4], acc[5], acc[6], acc[7]);
        *(float4*)(op)     = lo;   // 8 consecutive n per lane -> two b128 stores
        *(float4*)(op + 4) = hi;
    }
}

extern "C" void kernel_launch(void* const* d_in, const int* in_sizes, int n_in,
                              void* d_out, int out_size, void* d_ws, size_t ws_size,
                              hipStream_t stream) {
    (void)in_sizes; (void)n_in; (void)d_ws; (void)ws_size; (void)out_size;
    const float* e  = (const float*)d_in[0];   // (65536,)
    const float* W  = (const float*)d_in[1];   // (8,129)
    const float* b  = (const float*)d_in[2];   // (8,)
    const float* sc = (const float*)d_in[3];   // (1,)
    float* out = (float*)d_out;                // (8,65536,1)
    hermite_fused<<<NPTS / BLK, BLK, 0, stream>>>(e, W, b, sc, out);
}